// AxialAttention_40888088658298
// MI455X (gfx1250) — compile-verified
//
#include <hip/hip_runtime.h>
#include <hip/hip_bf16.h>

// ---------------------------------------------------------------------------
// Axial attention, MI455X (gfx1250), wave32, fp32 WMMA path.
//
// Logical problem after moveaxis/reshape: B=512 batches, L=512, D=64, fp32.
//   S = (Q K^T) * 1/8 ; mask; softmax; * head_mask ; out = P V
// d_out = out (16,777,216 f32, original permuted layout) ++ probs (512^3 f32).
//
// q/k/v/out memory layout (original (2,8,512,32,64), batch Bb = bh*32 + d2):
//   elem (Bb, l, d) at  bh*512*2048 + l*2048 + d2*64 + d      (floats)
// probs layout: contiguous (Bb, lq, lk).
// ---------------------------------------------------------------------------

typedef __attribute__((ext_vector_type(2))) float v2f;
typedef __attribute__((ext_vector_type(8))) float v8f;

#define L_SEQ     512
#define ROW_STR   2048              // 32*64 floats between sequence rows
#define OUT_ELEMS 16777216          // 2*8*512*32*64
#define PSTRIDE   516               // 512 + 4 pad (4 mod 64 banks -> conflict-free)
#define SCALE     0.125f            // 1/sqrt(64)
#define NEG_INF   -1e9f

__global__ __launch_bounds__(256)
void AxialAttention_40888088658298_kernel(const float* __restrict__ q,
                                          const float* __restrict__ k,
                                          const float* __restrict__ v,
                                          const float* __restrict__ amask,
                                          const float* __restrict__ hmask,
                                          float* __restrict__ out)
{
    const int qt   = blockIdx.x;          // q-tile: 16 rows, 0..31
    const int Bb   = blockIdx.y;          // batch:  0..511
    const int bh   = Bb >> 5;
    const int d2   = Bb & 31;
    const size_t base = (size_t)bh * L_SEQ * ROW_STR + (size_t)d2 * 64;

    const int lane = threadIdx.x & 31;
    const int wave = threadIdx.x >> 5;    // 0..7
    const int half = lane >> 4;           // 0/1 (lane-half -> K split / M split)
    const int ln   = lane & 15;

    __shared__ float redmax[128];
    __shared__ float redsum[128];
    __shared__ float Ptile[16 * PSTRIDE];       // 33 KB probs staging
    __shared__ float accred[4 * 8 * 32];        // 4 KB GEMM2 partials

    // ---- A fragments: Q tile 16x64, scale folded in -------------------------
    // A 16x4 f32 layout: lane l holds M=l%16, VGPR0/1 = K = 2*(l/16), +1.
    v2f aq[16];
    {
        const float* qr = q + base + (size_t)(qt * 16 + ln) * ROW_STR + half * 2;
        #pragma unroll
        for (int kk = 0; kk < 16; ++kk) {
            v2f t = *(const v2f*)(qr + kk * 4);
            aq[kk] = t * SCALE;
        }
    }

    // ---- GEMM1: S = Q K^T, this wave's 64-column slab (4 n-tiles) -----------
    // B 4x16 f32 layout: lane l holds N=l%16, VGPR0/1 = K = 2*(l/16), +1
    // -> contiguous float2 from row-major K at row N.
    const int nb0 = wave * 64;
    v8f sc[4];
    #pragma unroll
    for (int t = 0; t < 4; ++t) {
        v8f acc = {};
        const float* kr = k + base + (size_t)(nb0 + t * 16 + ln) * ROW_STR + half * 2;
        #pragma unroll
        for (int kk = 0; kk < 16; ++kk) {
            v2f bb = *(const v2f*)(kr + kk * 4);
            acc = __builtin_amdgcn_wmma_f32_16x16x4_f32(
                      false, aq[kk], false, bb, (short)0, acc, false, false);
        }
        sc[t] = acc;
    }

    // ---- attention mask (mask==0 -> NEG_INF) --------------------------------
    // C layout: VGPR r, lane l -> row M = r + 8*(l/16), col N = l%16.
    #pragma unroll
    for (int t = 0; t < 4; ++t) {
        const int kc = nb0 + t * 16 + ln;
        #pragma unroll
        for (int r = 0; r < 8; ++r) {
            const int qr = qt * 16 + r + 8 * half;
            float m = amask[qr * L_SEQ + kc];
            if (m == 0.0f) sc[t][r] = NEG_INF;
        }
    }

    // ---- row max: tiles -> 16-lane half -> cross-wave (LDS) -----------------
    float rm[8];
    #pragma unroll
    for (int r = 0; r < 8; ++r) {
        float m = fmaxf(fmaxf(sc[0][r], sc[1][r]), fmaxf(sc[2][r], sc[3][r]));
        #pragma unroll
        for (int off = 8; off >= 1; off >>= 1)
            m = fmaxf(m, __shfl_xor(m, off, 32));
        rm[r] = m;
    }
    if (ln == 0) {
        #pragma unroll
        for (int r = 0; r < 8; ++r)
            redmax[wave * 16 + half * 8 + r] = rm[r];
    }
    __syncthreads();
    float rmax[8];
    #pragma unroll
    for (int r = 0; r < 8; ++r) {
        float m = -3.0e38f;
        #pragma unroll
        for (int w = 0; w < 8; ++w)
            m = fmaxf(m, redmax[w * 16 + half * 8 + r]);
        rmax[r] = m;
    }

    // ---- exp + row sum ------------------------------------------------------
    float rs[8];
    #pragma unroll
    for (int r = 0; r < 8; ++r) rs[r] = 0.0f;
    #pragma unroll
    for (int t = 0; t < 4; ++t)
        #pragma unroll
        for (int r = 0; r < 8; ++r) {
            float e = __expf(sc[t][r] - rmax[r]);
            sc[t][r] = e;
            rs[r] += e;
        }
    #pragma unroll
    for (int r = 0; r < 8; ++r) {
        float s = rs[r];
        #pragma unroll
        for (int off = 8; off >= 1; off >>= 1)
            s += __shfl_xor(s, off, 32);
        rs[r] = s;
    }
    if (ln == 0) {
        #pragma unroll
        for (int r = 0; r < 8; ++r)
            redsum[wave * 16 + half * 8 + r] = rs[r];
    }
    __syncthreads();
    float rinv[8];
    #pragma unroll
    for (int r = 0; r < 8; ++r) {
        float s = 0.0f;
        #pragma unroll
        for (int w = 0; w < 8; ++w)
            s += redsum[w * 16 + half * 8 + r];
        rinv[r] = 1.0f / s;
    }

    // ---- probs = softmax * head_mask: write global + stage in LDS -----------
    float* probs = out + OUT_ELEMS + (size_t)Bb * L_SEQ * L_SEQ;
    #pragma unroll
    for (int t = 0; t < 4; ++t) {
        const int kc = nb0 + t * 16 + ln;
        #pragma unroll
        for (int r = 0; r < 8; ++r) {
            const int lr = r + 8 * half;            // local row 0..15
            const int qr = qt * 16 + lr;
            float p = sc[t][r] * rinv[r] * hmask[qr * L_SEQ + kc];
            probs[(size_t)qr * L_SEQ + kc] = p;
            Ptile[lr * PSTRIDE + kc] = p;
        }
    }
    __syncthreads();

    // ---- GEMM2: out(16x64) = P(16x512) @ V(512x64) --------------------------
    // wave -> n-tile (wave&3), K-half (wave>>2); partials combined via LDS.
    {
        const int nt    = wave & 3;
        const int khalf = wave >> 2;
        const int nc    = nt * 16 + ln;
        const float* vb = v + base + nc;
        v8f oacc = {};
        #pragma unroll 8
        for (int kk = khalf * 64; kk < khalf * 64 + 64; ++kk) {
            v2f pa = *(const v2f*)&Ptile[ln * PSTRIDE + kk * 4 + half * 2];
            const int kr = kk * 4 + half * 2;
            v2f bv = { vb[(size_t)kr * ROW_STR], vb[(size_t)(kr + 1) * ROW_STR] };
            oacc = __builtin_amdgcn_wmma_f32_16x16x4_f32(
                       false, pa, false, bv, (short)0, oacc, false, false);
        }
        if (wave >= 4) {
            #pragma unroll
            for (int r = 0; r < 8; ++r)
                accred[(wave - 4) * 256 + r * 32 + lane] = oacc[r];
        }
        __syncthreads();
        if (wave < 4) {
            #pragma unroll
            for (int r = 0; r < 8; ++r) {
                float o = oacc[r] + accred[wave * 256 + r * 32 + lane];
                const int orow = qt * 16 + r + 8 * half;
                out[base + (size_t)orow * ROW_STR + nc] = o;
            }
        }
    }
}

extern "C" void kernel_launch(void* const* d_in, const int* in_sizes, int n_in,
                              void* d_out, int out_size, void* d_ws, size_t ws_size,
                              hipStream_t stream) {
    const float* q  = (const float*)d_in[0];
    const float* k  = (const float*)d_in[1];
    const float* v  = (const float*)d_in[2];
    const float* am = (const float*)d_in[3];
    const float* hm = (const float*)d_in[4];
    float* out = (float*)d_out;

    dim3 grid(32, 512);   // (q-tile, batch)
    dim3 block(256);      // 8 wave32
    AxialAttention_40888088658298_kernel<<<grid, block, 0, stream>>>(q, k, v, am, hm, out);
}